// VGAE_83090437308748
// MI455X (gfx1250) — compile-verified
//
#include <hip/hip_runtime.h>
#include <hip/hip_bf16.h>
#include <math.h>

// Problem constants (match reference).
constexpr int NN   = 20000;  // nodes
constexpr int KK   = 16;     // neighbors per direction
constexpr int DD   = 256;    // embed dim
constexpr int TWOD = 512;    // 2*D
// LDS A-tile row stride in floats: 516 = multiple of 4 (16B-aligned rows for
// b128 staging) and 516 mod 64 = 4 -> 16-lane column reads hit banks
// 0,4,8,...,60 (all distinct, conflict-free).
constexpr int LDA  = 516;

constexpr int NKP      = 64;                      // k-pairs (each = 8 K values)
constexpr int WREP_ELEMS4 = 16 * NKP * 32;        // col-tiles * kpairs * lanes (float4)

typedef __attribute__((ext_vector_type(2))) float v2f;
typedef __attribute__((ext_vector_type(8))) float v8f;

// ---------------------------------------------------------------------------
// Aggregation: 4 nodes per block (64 threads/node, one float4 of D per thread).
// Gathered rows are read as coalesced b128 (64 lanes x 16B = one 1KB row).
// agg[n, 0:256]   = sum_k in_msk*nodes[in_idx] + out_msk*nodes[out_idx]
// agg[n, 256:512] = sum_k in_msk*E[in_edg]     + out_msk*E[out_edg]
// (in+out fused because both directions share the same W in the linear.)
// edgeOnly=1 recomputes only the edge half (gc3 reuses gc2's node half).
// ---------------------------------------------------------------------------
__global__ __launch_bounds__(256)
void agg_kernel(const float* __restrict__ nodes,
                const int*   __restrict__ inIdx,  const int*   __restrict__ inEdg,
                const float* __restrict__ inMsk,
                const int*   __restrict__ outIdx, const int*   __restrict__ outEdg,
                const float* __restrict__ outMsk,
                const float* __restrict__ E,
                float* __restrict__ agg,
                int edgeOnly)
{
    const int tid = threadIdx.x;
    const int sub = tid >> 6;          // node within block [0,4)
    const int t   = tid & 63;          // float4 slot within D  [0,64)
    const int n   = blockIdx.x * 4 + sub;

    __shared__ int   sIdx[4][2 * KK];
    __shared__ int   sEdg[4][2 * KK];
    __shared__ float sMsk[4][2 * KK];

    if (t < KK) {
        sIdx[sub][t] = inIdx[n * KK + t];
        sEdg[sub][t] = inEdg[n * KK + t];
        sMsk[sub][t] = inMsk[n * KK + t];
    } else if (t < 2 * KK) {
        const int j = t - KK;
        sIdx[sub][t] = outIdx[n * KK + j];
        sEdg[sub][t] = outEdg[n * KK + j];
        sMsk[sub][t] = outMsk[n * KK + j];
    }
    __syncthreads();

    const float4* __restrict__ nodes4 = (const float4*)nodes;   // rows of 64 float4
    const float4* __restrict__ E4     = (const float4*)E;
    float4*       __restrict__ agg4   = (float4*)agg;           // rows of 128 float4

    // Edge-embedding half (E is 64KB -> cache resident).
    {
        float4 e = make_float4(0.f, 0.f, 0.f, 0.f);
#pragma unroll
        for (int j = 0; j < 2 * KK; ++j) {
            const float  m = sMsk[sub][j];
            const float4 v = E4[(size_t)sEdg[sub][j] * 64 + t];
            e.x = fmaf(m, v.x, e.x);
            e.y = fmaf(m, v.y, e.y);
            e.z = fmaf(m, v.z, e.z);
            e.w = fmaf(m, v.w, e.w);
        }
        agg4[(size_t)n * 128 + 64 + t] = e;
    }

    if (!edgeOnly) {
        float4 s = make_float4(0.f, 0.f, 0.f, 0.f);
#pragma unroll
        for (int j = 0; j < 2 * KK; ++j) {
            const float  m = sMsk[sub][j];
            const float4 v = nodes4[(size_t)sIdx[sub][j] * 64 + t];
            s.x = fmaf(m, v.x, s.x);
            s.y = fmaf(m, v.y, s.y);
            s.z = fmaf(m, v.z, s.z);
            s.w = fmaf(m, v.w, s.w);
        }
        agg4[(size_t)n * 128 + t] = s;
    }
}

// ---------------------------------------------------------------------------
// Repack W [512,256] into WMMA B-fragment order:
//   Wrep4[((ct*64 + kp)*32 + lane)] = float4(
//       W[(8kp + hk  )*256 + c], W[(8kp + hk+1)*256 + c],    // k-step 2kp
//       W[(8kp+4 + hk)*256 + c], W[(8kp+4+hk+1)*256 + c])    // k-step 2kp+1
//   with c = ct*16 + (lane&15), hk = (lane>=16)?2:0.
// In the GEMM, a wave's B loads become lane-consecutive b128 (512B bursts).
// One-shot: 512KB, reused by all 1250 GEMM blocks.
// ---------------------------------------------------------------------------
__global__ __launch_bounds__(256)
void repack_w_kernel(const float* __restrict__ W, float4* __restrict__ Wrep4)
{
    const int gid = blockIdx.x * 256 + threadIdx.x;   // [0, 32768)
    const int l   = gid & 31;
    const int kp  = (gid >> 5) & (NKP - 1);
    const int ct  = gid >> 11;
    const int c   = ct * 16 + (l & 15);
    const int hk  = (l >= 16) ? 2 : 0;
    const int r0  = 8 * kp + hk;
    Wrep4[gid] = make_float4(W[(size_t)(r0    ) * DD + c],
                             W[(size_t)(r0 + 1) * DD + c],
                             W[(size_t)(r0 + 4) * DD + c],
                             W[(size_t)(r0 + 5) * DD + c]);
}

// ---------------------------------------------------------------------------
// WMMA GEMM: y[16x256 tile] = prev + agg(16x512) @ W(512x256) + 2*bias
// 512 threads = 16 wave32s; wave w owns output columns [16w, 16w+16).
// A tile staged in LDS via b128; B fragments streamed from pre-packed Wrep
// as coalesced b128. kldMode=1: fuse the KLD term reduction instead of
// storing y (logvar is never materialized).
// ---------------------------------------------------------------------------
__global__ __launch_bounds__(512)
void gemm_wmma_kernel(const float* __restrict__ agg,
                      const float4* __restrict__ Wrep4,
                      const float* __restrict__ bias,
                      const float* __restrict__ prev,
                      float* __restrict__ out,
                      const float* __restrict__ mu,
                      float* __restrict__ kldAcc,
                      int kldMode)
{
    __shared__ float As[16 * LDA];
    __shared__ float wsum[16];

    const int tid     = threadIdx.x;
    const int rowBase = blockIdx.x * 16;   // N = 1250 * 16 exactly, no tails

    // Stage the 16x512 A tile: 2048 float4s, 4 b128 loads + ds_store_b128 each.
    {
        const float4* __restrict__ agg4 = (const float4*)agg;
#pragma unroll
        for (int i = 0; i < 4; ++i) {
            const int idx4 = tid + i * 512;       // [0, 2048)
            const int r    = idx4 >> 7;           // 128 float4 per row
            const int c4   = (idx4 & 127) * 4;    // float column
            const float4 v = agg4[(size_t)(rowBase + r) * 128 + (idx4 & 127)];
            *(float4*)&As[r * LDA + c4] = v;
        }
    }
    __syncthreads();

    const int wave  = tid >> 5;
    const int lane  = tid & 31;
    const int col0  = wave * 16;
    const int mrow  = lane & 15;            // matrix row (A) / column offset (B)
    const int halfk = (lane >= 16) ? 2 : 0; // ISA 16x4 f32 A layout: hi half-wave = K+2

    const float*  __restrict__ aRow = &As[mrow * LDA];
    const float4* __restrict__ wp4  = Wrep4 + ((size_t)wave * NKP) * 32 + lane;

    v8f acc = {};
#pragma unroll 4
    for (int kp = 0; kp < NKP; ++kp) {
        const float4 bb = wp4[(size_t)kp * 32];   // lane-consecutive b128 stream
        const int    k0 = kp * 8;
        v2f a0, a1, b0, b1;
        a0.x = aRow[k0 + halfk];
        a0.y = aRow[k0 + halfk + 1];
        a1.x = aRow[k0 + 4 + halfk];
        a1.y = aRow[k0 + 4 + halfk + 1];
        b0.x = bb.x; b0.y = bb.y;
        b1.x = bb.z; b1.y = bb.w;
        acc = __builtin_amdgcn_wmma_f32_16x16x4_f32(
            false, a0, false, b0, (short)0, acc, false, false);
        acc = __builtin_amdgcn_wmma_f32_16x16x4_f32(
            false, a1, false, b1, (short)0, acc, false, false);
    }

    // C/D layout: VGPR v holds row (v + 8*hi), column = col0 + mrow.
    const int   col   = col0 + mrow;
    const int   rOff  = (lane >= 16) ? 8 : 0;
    const float bias2 = 2.0f * bias[col];   // bias added for both in and out linears

    if (!kldMode) {
#pragma unroll
        for (int v = 0; v < 8; ++v) {
            const int row = rowBase + v + rOff;
            out[(size_t)row * DD + col] = prev[(size_t)row * DD + col] + acc[v] + bias2;
        }
    } else {
        float part = 0.f;
#pragma unroll
        for (int v = 0; v < 8; ++v) {
            const int   row = rowBase + v + rOff;
            const float lv  = prev[(size_t)row * DD + col] + acc[v] + bias2; // logvar
            const float lvT = tanhf(lv);
            const float muT = tanhf(mu[(size_t)row * DD + col]);
            part += 1.0f + 2.0f * lvT - muT * muT - __expf(2.0f * lvT);
        }
        // wave32 reduce
#pragma unroll
        for (int off = 16; off > 0; off >>= 1)
            part += __shfl_down(part, off, 32);
        if (lane == 0) wsum[wave] = part;
        __syncthreads();
        if (tid < 16) {
            float s = wsum[tid];
#pragma unroll
            for (int off = 8; off > 0; off >>= 1)
                s += __shfl_down(s, off, 16);
            if (tid == 0) atomicAdd(kldAcc, s);
        }
    }
}

__global__ void init_acc_kernel(float* __restrict__ acc)
{
    if (threadIdx.x == 0) *acc = 0.0f;
}

__global__ void finalize_kld_kernel(const float* __restrict__ acc,
                                    float* __restrict__ outScalar)
{
    if (threadIdx.x == 0) {
        // kld = (-0.5/N) * mean_n( sum_d term ) = -0.5 * total / N^2
        *outScalar = -0.5f * (*acc) / ((float)NN * (float)NN);
    }
}

// ---------------------------------------------------------------------------
extern "C" void kernel_launch(void* const* d_in, const int* in_sizes, int n_in,
                              void* d_out, int out_size, void* d_ws, size_t ws_size,
                              hipStream_t stream)
{
    (void)in_sizes; (void)n_in; (void)out_size; (void)ws_size;

    const float* node_reps = (const float*)d_in[0];
    const int*   in_idx    = (const int*)  d_in[1];
    const int*   in_edg    = (const int*)  d_in[2];
    const float* in_msk    = (const float*)d_in[3];
    const int*   out_idx   = (const int*)  d_in[4];
    const int*   out_edg   = (const int*)  d_in[5];
    const float* out_msk   = (const float*)d_in[6];
    const float* E1 = (const float*)d_in[7];
    const float* W1 = (const float*)d_in[8];
    const float* b1 = (const float*)d_in[9];
    const float* E2 = (const float*)d_in[10];
    const float* W2 = (const float*)d_in[11];
    const float* b2 = (const float*)d_in[12];
    const float* E3 = (const float*)d_in[13];
    const float* W3 = (const float*)d_in[14];
    const float* b3 = (const float*)d_in[15];

    float* out    = (float*)d_out;
    float* hidden = out;                      // [N, D] lives directly in d_out
    float* kldOut = out + (size_t)NN * DD;    // trailing scalar

    float*  agg    = (float*)d_ws;                      // [N, 2D]
    float*  mu     = agg + (size_t)NN * TWOD;           // [N, D]
    float4* Wrep   = (float4*)(mu + (size_t)NN * DD);   // 512KB, 16B-aligned
    float*  kldAcc = (float*)(Wrep + WREP_ELEMS4);      // 1 float

    const int aggBlocks  = NN / 4;            // 5000 (4 nodes per block)
    const int gemmBlocks = NN / 16;           // 1250
    const int repBlocks  = WREP_ELEMS4 / 256; // 128

    init_acc_kernel<<<1, 64, 0, stream>>>(kldAcc);

    // gc1: hidden = node_reps + agg(node_reps, E1) @ W1 + 2*b1
    repack_w_kernel<<<repBlocks, 256, 0, stream>>>(W1, Wrep);
    agg_kernel<<<aggBlocks, 256, 0, stream>>>(node_reps, in_idx, in_edg, in_msk,
                                              out_idx, out_edg, out_msk, E1, agg, 0);
    gemm_wmma_kernel<<<gemmBlocks, 512, 0, stream>>>(agg, Wrep, b1, node_reps,
                                                     hidden, nullptr, nullptr, 0);

    // gc2: mu = hidden + agg(hidden, E2) @ W2 + 2*b2
    repack_w_kernel<<<repBlocks, 256, 0, stream>>>(W2, Wrep);
    agg_kernel<<<aggBlocks, 256, 0, stream>>>(hidden, in_idx, in_edg, in_msk,
                                              out_idx, out_edg, out_msk, E2, agg, 0);
    gemm_wmma_kernel<<<gemmBlocks, 512, 0, stream>>>(agg, Wrep, b2, hidden,
                                                     mu, nullptr, nullptr, 0);

    // gc3: node half of agg identical to gc2's -> recompute edge half only.
    repack_w_kernel<<<repBlocks, 256, 0, stream>>>(W3, Wrep);
    agg_kernel<<<aggBlocks, 256, 0, stream>>>(hidden, in_idx, in_edg, in_msk,
                                              out_idx, out_edg, out_msk, E3, agg, 1);
    // logvar never materialized: fused KLD reduction in the GEMM epilogue.
    gemm_wmma_kernel<<<gemmBlocks, 512, 0, stream>>>(agg, Wrep, b3, hidden,
                                                     nullptr, mu, kldAcc, 1);

    finalize_kld_kernel<<<1, 64, 0, stream>>>(kldAcc, kldOut);
}